// LindBladEvolve_41137196761438
// MI455X (gfx1250) — compile-verified
//
#include <hip/hip_runtime.h>

#define DIM 16
#define BATCH 32
#define K_LIND 4
#define N_TS 512
#define N_CTRLS 4
#define N_PARAMS 16
#define DEGREE 3
#define DTSTEP (1.0f / 512.0f)

typedef float v2f __attribute__((ext_vector_type(2)));
typedef float v8f __attribute__((ext_vector_type(8)));

// ---------------- workspace layout (float offsets) ----------------
// u      : 512*4   = 2048
// basere : 256
// baseim : 256
// Ccre   : 4*256   = 1024
// Ccim   : 4*256   = 1024
// Gre    : 512*256 = 131072
// Gim    : 512*256 = 131072
#define WS_U      0
#define WS_BASERE 2048
#define WS_BASEIM 2304
#define WS_CCRE   2560
#define WS_CCIM   3584
#define WS_GRE    4608
#define WS_GIM    135680
// total 266752 floats = ~1.02 MB

// ======================= kernel 1: pulses + static matrices =======================
__global__ __launch_bounds__(512) void lind_prep(
    const float* __restrict__ params,
    const float* __restrict__ H0_re, const float* __restrict__ H0_im,
    const float* __restrict__ Hc_re, const float* __restrict__ Hc_im,
    const float* __restrict__ L_re,  const float* __restrict__ L_im,
    float* __restrict__ ws)
{
  const int tid = threadIdx.x;

  // ---- B-spline control pulses: thread t computes u[t][0..3] ----
  {
    const float tv = tid * DTSTEP;
    float kn[20];
    kn[0] = kn[1] = kn[2] = 0.0f;
#pragma unroll
    for (int i = 0; i < 14; ++i) kn[3 + i] = (float)i / 13.0f;
    kn[17] = kn[18] = kn[19] = 1.0f;

    float Bv[19];
#pragma unroll
    for (int i = 0; i < 19; ++i)
      Bv[i] = (kn[i] <= tv && tv < kn[i + 1]) ? 1.0f : 0.0f;

#pragma unroll
    for (int d = 1; d <= 3; ++d) {
      const int m = 19 - d;
      for (int i = 0; i < m; ++i) {
        const float ld = kn[i + d] - kn[i];
        const float rd = kn[i + d + 1] - kn[i + 1];
        const float lt = (ld > 0.0f) ? (tv - kn[i]) / ld * Bv[i] : 0.0f;
        const float rt = (rd > 0.0f) ? (kn[i + d + 1] - tv) / rd * Bv[i + 1] : 0.0f;
        Bv[i] = lt + rt;
      }
    }
#pragma unroll
    for (int c = 0; c < N_CTRLS; ++c) {
      float acc = 0.0f;
#pragma unroll
      for (int p = 0; p < N_PARAMS; ++p) acc += Bv[p] * params[p * N_CTRLS + c];
      ws[WS_U + tid * N_CTRLS + c] = acc;
    }
  }

  // ---- static matrices: thread e (<256) computes element (r,c) ----
  if (tid < 256) {
    const int r = tid >> 4, c = tid & 15;

    // Hermitized H0
    const float h0re = 0.5f * (H0_re[r * 16 + c] + H0_re[c * 16 + r]);
    const float h0im = 0.5f * (H0_im[r * 16 + c] - H0_im[c * 16 + r]);

    // LdL[r][c] = sum_k sum_j conj(L[k][j][r]) * L[k][j][c]
    float ldl_re = 0.0f, ldl_im = 0.0f;
    for (int k = 0; k < K_LIND; ++k) {
      const float* lre = L_re + k * 256;
      const float* lim = L_im + k * 256;
      for (int j = 0; j < 16; ++j) {
        const float ar = lre[j * 16 + r], ai = lim[j * 16 + r];
        const float br = lre[j * 16 + c], bi = lim[j * 16 + c];
        ldl_re += ar * br + ai * bi;
        ldl_im += ar * bi - ai * br;
      }
    }
    // base = -i*H0h - 0.5*LdL
    ws[WS_BASERE + tid] =  h0im - 0.5f * ldl_re;
    ws[WS_BASEIM + tid] = -h0re - 0.5f * ldl_im;

    // Cc = -i * Hermitize(Hc[cc])
#pragma unroll
    for (int cc = 0; cc < N_CTRLS; ++cc) {
      const float* hre = Hc_re + cc * 256;
      const float* him = Hc_im + cc * 256;
      const float hhre = 0.5f * (hre[r * 16 + c] + hre[c * 16 + r]);
      const float hhim = 0.5f * (him[r * 16 + c] - him[c * 16 + r]);
      ws[WS_CCRE + cc * 256 + tid] =  hhim;
      ws[WS_CCIM + cc * 256 + tid] = -hhre;
    }
  }
}

// ======================= kernel 2: Geff(t) table =======================
__global__ __launch_bounds__(256) void lind_gbuild(float* __restrict__ ws)
{
  const int t = blockIdx.x;   // 0..511
  const int e = threadIdx.x;  // 0..255
  float re = ws[WS_BASERE + e];
  float im = ws[WS_BASEIM + e];
#pragma unroll
  for (int c = 0; c < N_CTRLS; ++c) {
    const float uc = ws[WS_U + t * N_CTRLS + c];
    re += uc * ws[WS_CCRE + c * 256 + e];
    im += uc * ws[WS_CCIM + c * 256 + e];
  }
  ws[WS_GRE + t * 256 + e] = re;
  ws[WS_GIM + t * 256 + e] = im;
}

// ======================= WMMA helpers =======================
__device__ __forceinline__ v8f wmma4(v2f a, v2f b, v8f c) {
  // D = A(16x4,f32) * B(4x16,f32) + C(16x16,f32)
  return __builtin_amdgcn_wmma_f32_16x16x4_f32(false, a, false, b, (short)0, c,
                                               false, false);
}

// Complex 16x16 matmul: D = A * B, A in A-layout frags, B in B-layout frags.
// Bin must hold -Bi (f32 WMMA has no A/B negate modifier per ISA NEG table).
__device__ __forceinline__ void cmm(const v2f* Ar, const v2f* Ai,
                                    const v2f* Br, const v2f* Bi, const v2f* Bin,
                                    v8f& Dr, v8f& Di)
{
  v8f cr = {}; v8f ci = {};
#pragma unroll
  for (int s = 0; s < 4; ++s) {
    cr = wmma4(Ar[s], Br[s], cr);   // + Ar*Br
    cr = wmma4(Ai[s], Bin[s], cr);  // - Ai*Bi
    ci = wmma4(Ar[s], Bi[s], ci);   // + Ar*Bi
    ci = wmma4(Ai[s], Br[s], ci);   // + Ai*Br
  }
  Dr = cr; Di = ci;
}

// ======================= kernel 3: Euler evolution =======================
__global__ __launch_bounds__(128) void lind_evolve(
    const float* __restrict__ L_re, const float* __restrict__ L_im,
    const float* __restrict__ rho0_re, const float* __restrict__ rho0_im,
    const float* __restrict__ ws, float* __restrict__ out)
{
  __shared__ float rho[512];        // [0..255] re, [256..511] im (row-major 16x16)
  __shared__ float part[4][512];    // per-wave drho partials, same layout
  __shared__ float scr[4][512];     // per-wave scratch for T = L_k * rho

  const int tid  = threadIdx.x;
  const int w    = tid >> 5;        // wave id 0..3 (== Lindblad index k)
  const int lane = tid & 31;
  const int b    = blockIdx.x;      // batch element

  const int row  = lane & 15;
  const int col  = lane & 15;
  const int koff = (lane >> 4) << 1;    // 0 for lanes 0-15, 2 for lanes 16-31
  const int rhalf = (lane >> 4) << 3;   // D-frag row offset: 0 or 8

  // ---- load initial rho into LDS ----
#pragma unroll
  for (int j = 0; j < 4; ++j) {
    const int idx = tid + j * 128;
    rho[idx] = (idx < 256) ? rho0_re[b * 256 + idx] : rho0_im[b * 256 + (idx - 256)];
  }

  // ---- constant fragments for this wave: L_w (A-layout) and L_w^dag (B-layout) ----
  const float* lre = L_re + w * 256;
  const float* lim = L_im + w * 256;
  v2f LAre[4], LAim[4], LdBre[4], LdBim[4], LdBin[4];
#pragma unroll
  for (int s = 0; s < 4; ++s) {
    const int kb = 4 * s + koff;
    LAre[s] = (v2f){ lre[row * 16 + kb], lre[row * 16 + kb + 1] };
    LAim[s] = (v2f){ lim[row * 16 + kb], lim[row * 16 + kb + 1] };
    // Ldag[kb][col] = conj(L[col][kb])
    LdBre[s] = (v2f){  lre[col * 16 + kb],  lre[col * 16 + kb + 1] };
    LdBim[s] = (v2f){ -lim[col * 16 + kb], -lim[col * 16 + kb + 1] };
    LdBin[s] = -LdBim[s];
  }

  const float* Gre = ws + WS_GRE;
  const float* Gim = ws + WS_GIM;

  for (int t = 0; t < N_TS; ++t) {
    // ---- issue G(t) loads BEFORE the barrier: they depend only on t, not rho,
    //      so their L2 latency overlaps the barrier wait + rho fragment loads ----
    v2f GAre[4], GAim[4];            // wave 0: G(t) in A-layout
    v2f GdBre[4], GdBim[4], GdBin[4]; // wave 1: G(t)^dag in B-layout
    if (w == 0) {
      const float* gr = Gre + t * 256;
      const float* gi = Gim + t * 256;
#pragma unroll
      for (int s = 0; s < 4; ++s) {
        const int kb = 4 * s + koff;
        GAre[s] = (v2f){ gr[row * 16 + kb], gr[row * 16 + kb + 1] };
        GAim[s] = (v2f){ gi[row * 16 + kb], gi[row * 16 + kb + 1] };
      }
    } else if (w == 1) {
      const float* gr = Gre + t * 256;
      const float* gi = Gim + t * 256;
#pragma unroll
      for (int s = 0; s < 4; ++s) {
        const int kb = 4 * s + koff;
        // Gdag[kb][col] = conj(G[col][kb])
        GdBre[s] = (v2f){  gr[col * 16 + kb],  gr[col * 16 + kb + 1] };
        GdBim[s] = (v2f){ -gi[col * 16 + kb], -gi[col * 16 + kb + 1] };
        GdBin[s] = -GdBim[s];
      }
    }
    // prefetch next step's G row into near cache (global_prefetch_b8);
    // clamp so the last iteration stays inside mapped workspace
    if (w < 2) {
      const int t1 = (t < N_TS - 1) ? (t + 1) : t;
      __builtin_prefetch(Gre + t1 * 256 + lane * 8, 0, 3);
      __builtin_prefetch(Gim + t1 * 256 + lane * 8, 0, 3);
    }

    __syncthreads();  // rho stable before fragment loads

    // ---- rho B-layout fragments (needed by every wave) ----
    v2f rBre[4], rBim[4], rBin[4];
#pragma unroll
    for (int s = 0; s < 4; ++s) {
      const int kb = 4 * s + koff;
      rBre[s] = (v2f){ rho[kb * 16 + col],       rho[(kb + 1) * 16 + col] };
      rBim[s] = (v2f){ rho[256 + kb * 16 + col], rho[256 + (kb + 1) * 16 + col] };
      rBin[s] = -rBim[s];
    }

    // ---- T = L_w * rho ----
    v8f Tr, Ti;
    cmm(LAre, LAim, rBre, rBim, rBin, Tr, Ti);

    // bounce T through per-wave scratch to re-fragment as A-layout
#pragma unroll
    for (int r = 0; r < 8; ++r) {
      const int rr = r + rhalf;
      scr[w][rr * 16 + col]       = Tr[r];
      scr[w][256 + rr * 16 + col] = Ti[r];
    }
    v2f TAre[4], TAim[4];
#pragma unroll
    for (int s = 0; s < 4; ++s) {
      const int kb = 4 * s + koff;
      TAre[s] = (v2f){ scr[w][row * 16 + kb],       scr[w][row * 16 + kb + 1] };
      TAim[s] = (v2f){ scr[w][256 + row * 16 + kb], scr[w][256 + row * 16 + kb + 1] };
    }

    // ---- J = T * L_w^dag  (this wave's partial) ----
    v8f Pr, Pi;
    cmm(TAre, TAim, LdBre, LdBim, LdBin, Pr, Pi);

    if (w == 0) {
      // P += G(t) * rho
      v8f Qr, Qi;
      cmm(GAre, GAim, rBre, rBim, rBin, Qr, Qi);
      Pr += Qr; Pi += Qi;
    } else if (w == 1) {
      // P += rho * G(t)^dag  (rho as A-layout)
      v2f rAre[4], rAim[4];
#pragma unroll
      for (int s = 0; s < 4; ++s) {
        const int kb = 4 * s + koff;
        rAre[s] = (v2f){ rho[row * 16 + kb],       rho[row * 16 + kb + 1] };
        rAim[s] = (v2f){ rho[256 + row * 16 + kb], rho[256 + row * 16 + kb + 1] };
      }
      v8f Qr, Qi;
      cmm(rAre, rAim, GdBre, GdBim, GdBin, Qr, Qi);
      Pr += Qr; Pi += Qi;
    }

    // ---- store this wave's partial (row-major via D-frag mapping) ----
#pragma unroll
    for (int r = 0; r < 8; ++r) {
      const int rr = r + rhalf;
      part[w][rr * 16 + col]       = Pr[r];
      part[w][256 + rr * 16 + col] = Pi[r];
    }

    __syncthreads();  // partials visible

    // ---- rho += DT * (p0+p1+p2+p3); emit diag populations ----
#pragma unroll
    for (int j = 0; j < 4; ++j) {
      const int idx = tid + j * 128;
      const float s4 = part[0][idx] + part[1][idx] + part[2][idx] + part[3][idx];
      const float nv = rho[idx] + DTSTEP * s4;
      rho[idx] = nv;
      if (idx < 256 && (idx & 15) == (idx >> 4)) {
        out[t * (BATCH * DIM) + b * DIM + (idx >> 4)] = nv;
      }
    }
    // next iteration's top barrier protects update -> reload
  }
}

// ======================= launcher =======================
extern "C" void kernel_launch(void* const* d_in, const int* in_sizes, int n_in,
                              void* d_out, int out_size, void* d_ws, size_t ws_size,
                              hipStream_t stream)
{
  const float* params  = (const float*)d_in[0];
  const float* H0_re   = (const float*)d_in[1];
  const float* H0_im   = (const float*)d_in[2];
  const float* Hc_re   = (const float*)d_in[3];
  const float* Hc_im   = (const float*)d_in[4];
  const float* L_re    = (const float*)d_in[5];
  const float* L_im    = (const float*)d_in[6];
  const float* rho0_re = (const float*)d_in[7];
  const float* rho0_im = (const float*)d_in[8];
  float* out = (float*)d_out;
  float* ws  = (float*)d_ws;

  lind_prep<<<1, 512, 0, stream>>>(params, H0_re, H0_im, Hc_re, Hc_im,
                                   L_re, L_im, ws);
  lind_gbuild<<<N_TS, 256, 0, stream>>>(ws);
  lind_evolve<<<BATCH, 128, 0, stream>>>(L_re, L_im, rho0_re, rho0_im, ws, out);
}